// Post_process_deconv_67800353735285
// MI455X (gfx1250) — compile-verified
//
#include <hip/hip_runtime.h>
#include <stdint.h>

// Modulated deformable conv (deform_conv2d, 1ch, K=3, pad=1) for MI455X.
// Memory-bound: ~243 MB streaming -> ~10.4us roofline @ 23.3 TB/s.
// Depth tile staged into LDS with CDNA5 async global->LDS B128 DMA
// (ASYNCcnt / s_wait_asynccnt), zero-padded for image borders; bilinear
// gathers served from LDS with an exact global-memory fallback for the
// rare far-offset samples. Offset/mask streams and output use NT hints.

#define KK    9
#define TW    64
#define TH    8
#define HALO  8
#define LDSW  (TW + 2 * HALO)   // 80 floats per tile row
#define LDSH  (TH + 2 * HALO)   // 24 tile rows
#define CHUNKS_PER_ROW (LDSW / 4)   // 20 float4 chunks
#define NTHREADS 256

__device__ __forceinline__ float corner_global(const float* __restrict__ img,
                                               int H, int W, int yc, int xc) {
    if (yc < 0 || yc >= H || xc < 0 || xc >= W) return 0.0f;
    return img[(size_t)yc * W + xc];
}

__global__ __launch_bounds__(NTHREADS)
void deform_conv_kernel(const float* __restrict__ depth,   // [B,1,H,W]
                        const float* __restrict__ mask,    // [B,9,H,W]
                        const float* __restrict__ offs,    // [B,18,H,W]
                        const float* __restrict__ wk,      // [9]
                        const float* __restrict__ bias,    // [1]
                        float* __restrict__ out,           // [B,1,H,W]
                        int H, int W) {
    __shared__ __align__(16) float tile[LDSH * LDSW];

    const int b   = blockIdx.z;
    const int bx0 = blockIdx.x * TW;
    const int by0 = blockIdx.y * TH;
    const int tx0 = bx0 - HALO;      // multiple of 4 (bx0 mult of 64, HALO=8)
    const int ty0 = by0 - HALO;
    const int HWp = H * W;
    const float* __restrict__ img = depth + (size_t)b * HWp;

    const int tid = threadIdx.x;

    // ---- Stage depth tile into LDS in 16-byte chunks. ----
    // W is a multiple of 4 and tx0 is a multiple of 4, so each float4 chunk
    // of a tile row is entirely inside or entirely outside the image in x:
    // in-image chunks -> async global->LDS B128 DMA (16B-aligned both sides),
    // out-of-image chunks -> float4 zero store (disjoint LDS addresses).
    for (int i = tid; i < LDSH * CHUNKS_PER_ROW; i += NTHREADS) {
        const int r  = i / CHUNKS_PER_ROW;
        const int c4 = i - r * CHUNKS_PER_ROW;
        const int yy = ty0 + r;
        const int xx = tx0 + c4 * 4;
        float* ldst = &tile[r * LDSW + c4 * 4];
        if (yy >= 0 && yy < H && xx >= 0 && (xx + 3) < W) {
            const float* gp = img + (size_t)yy * W + xx;
            const uint32_t lds_off = (uint32_t)(size_t)ldst;  // low 32b of flat = LDS byte addr
            const uint64_t gaddr   = (uint64_t)(size_t)gp;
            asm volatile("global_load_async_to_lds_b128 %0, %1, off"
                         :: "v"(lds_off), "v"(gaddr)
                         : "memory");
        } else {
            *(float4*)ldst = make_float4(0.0f, 0.0f, 0.0f, 0.0f);
        }
    }
    // Each wave drains its own async DMAs; barrier publishes LDS to all waves.
    asm volatile("s_wait_asynccnt 0" ::: "memory");
    __syncthreads();

    // ---- Per-thread constants ----
    float wreg[KK];
#pragma unroll
    for (int k = 0; k < KK; ++k) wreg[k] = wk[k];
    const float bval = bias[0];

    const float* __restrict__ maskB = mask + (size_t)b * KK * HWp;
    const float* __restrict__ offB  = offs + (size_t)b * (2 * KK) * HWp;

    const int lx  = tid & (TW - 1);  // 0..63
    const int lyq = tid >> 6;        // 0..3

#pragma unroll
    for (int half = 0; half < 2; ++half) {
        const int ly  = lyq + half * 4;
        const int oy  = by0 + ly;
        const int ox  = bx0 + lx;
        const int pix = oy * W + ox;

        float acc = bval;
#pragma unroll
        for (int k = 0; k < KK; ++k) {
            const int ky = k / 3;
            const int kx = k - 3 * ky;
            // Single-use streams: non-temporal, keep cache for the depth image.
            const float dy = __builtin_nontemporal_load(&offB[(size_t)(2 * k)     * HWp + pix]);
            const float dx = __builtin_nontemporal_load(&offB[(size_t)(2 * k + 1) * HWp + pix]);
            const float m  = __builtin_nontemporal_load(&maskB[(size_t)k * HWp + pix]);

            const float y = (float)(oy - 1 + ky) + dy;
            const float x = (float)(ox - 1 + kx) + dx;
            const float y0f = floorf(y);
            const float x0f = floorf(x);
            const float wy = y - y0f;
            const float wx = x - x0f;
            const int y0 = (int)y0f;
            const int x0 = (int)x0f;
            const int y1 = y0 + 1;
            const int x1 = x0 + 1;

            float v00, v01, v10, v11;
            const int ry = y0 - ty0;
            const int rx = x0 - tx0;
            if (ry >= 0 && (ry + 1) < LDSH && rx >= 0 && (rx + 1) < LDSW) {
                // Fast path: all 4 corners inside the zero-padded LDS tile.
                const float* t = &tile[ry * LDSW + rx];
                v00 = t[0];
                v01 = t[1];
                v10 = t[LDSW];
                v11 = t[LDSW + 1];
            } else {
                // Rare tail (|offset| > ~6 sigma) and far borders: exact global path.
                v00 = corner_global(img, H, W, y0, x0);
                v01 = corner_global(img, H, W, y0, x1);
                v10 = corner_global(img, H, W, y1, x0);
                v11 = corner_global(img, H, W, y1, x1);
            }

            const float s = v00 * (1.0f - wy) * (1.0f - wx)
                          + v01 * (1.0f - wy) * wx
                          + v10 * wy * (1.0f - wx)
                          + v11 * wy * wx;
            acc = fmaf(s * m, wreg[k], acc);
        }
        __builtin_nontemporal_store(acc, &out[(size_t)b * HWp + pix]);
    }
}

extern "C" void kernel_launch(void* const* d_in, const int* in_sizes, int n_in,
                              void* d_out, int out_size, void* d_ws, size_t ws_size,
                              hipStream_t stream) {
    const float* depth = (const float*)d_in[0];  // [B,1,H,W]
    const float* mask  = (const float*)d_in[1];  // [B,9,H,W]   ("weight" in setup)
    const float* offs  = (const float*)d_in[2];  // [B,18,H,W]
    const float* wk    = (const float*)d_in[3];  // [1,1,3,3]
    const float* bias  = (const float*)d_in[4];  // [1]
    float* out = (float*)d_out;

    const int H = 512, W = 512;
    const int B = in_sizes[0] / (H * W);

    dim3 grid(W / TW, H / TH, B);   // 8 x 64 x 8
    dim3 block(NTHREADS);
    deform_conv_kernel<<<grid, block, 0, stream>>>(depth, mask, offs, wk, bias, out, H, W);
}